// StabilizedCrossAttention_81638738362592
// MI455X (gfx1250) — compile-verified
//
#include <hip/hip_runtime.h>
#include <hip/hip_bf16.h>
#include <math.h>

// ---------------------------------------------------------------------------
// StabilizedCrossAttention for MI455X (gfx1250, wave32, WMMA)
// B=4, Q=1024, K=2048, HS=1024, NH=16, HD=64, MAX_REL=512, WINDOW=5
// fp32 end-to-end via V_WMMA_F32_16X16X4_F32 (precision-safe vs fp32 ref).
// ---------------------------------------------------------------------------

typedef __attribute__((ext_vector_type(2))) float v2f;
typedef __attribute__((ext_vector_type(8))) float v8f;

#define HS   1024
#define NH   16
#define HD   64
#define QLEN 1024
#define KLEN 2048
#define BSZ  4

static __device__ __forceinline__ v8f wmma_f32(v2f a, v2f b, v8f c) {
  // 8 args: (neg_a, A, neg_b, B, c_mod, C, reuse_a, reuse_b)
  return __builtin_amdgcn_wmma_f32_16x16x4_f32(false, a, false, b, (short)0, c,
                                               false, false);
}

// ---------------------------------------------------------------------------
// proj_gemm: Y[m,n] = sum_h X[m,h] * W[n,h] + bias[n]     (i.e. X @ W^T + b)
// Block: 256 threads (8 waves). Block tile 128(M) x 64(N), K staged 32-deep
// in LDS. Wave w computes rows 16w..16w+15 across all 64 N columns.
// ---------------------------------------------------------------------------
#define TM 128
#define TN 64
#define TK 32
#define PITCH 36  // 36*l mod 64 spreads 16 lanes over 16 distinct LDS banks

__global__ __launch_bounds__(256) void proj_gemm(
    const float* __restrict__ X, const float* __restrict__ W,
    const float* __restrict__ bias, float* __restrict__ Y) {
  __shared__ float As[TM * PITCH];
  __shared__ float Ws[TN * PITCH];

  const int t    = threadIdx.x;
  const int wave = t >> 5;
  const int lane = t & 31;
  const int l16  = lane & 15;
  const int hi   = lane >> 4;               // half-wave select (K 0/1 vs 2/3)
  const int m0   = blockIdx.x * TM;
  const int n0   = blockIdx.y * TN;

  v8f acc[4];
#pragma unroll
  for (int j = 0; j < 4; ++j) acc[j] = (v8f)0.0f;

  const int ldr = t >> 3;        // 0..31
  const int ldc = (t & 7) * 4;   // 0..28 step 4

  for (int k0 = 0; k0 < HS; k0 += TK) {
    // ---- stage X tile [128 x 32] (coalesced float4) ----
#pragma unroll
    for (int i = 0; i < 4; ++i) {
      const int row = ldr + 32 * i;
      const float4 v =
          *(const float4*)(X + (size_t)(m0 + row) * HS + k0 + ldc);
      float* d = &As[row * PITCH + ldc];
      d[0] = v.x; d[1] = v.y; d[2] = v.z; d[3] = v.w;
    }
    // ---- stage W tile [64 x 32] ----
#pragma unroll
    for (int i = 0; i < 2; ++i) {
      const int row = ldr + 32 * i;
      const float4 v =
          *(const float4*)(W + (size_t)(n0 + row) * HS + k0 + ldc);
      float* d = &Ws[row * PITCH + ldc];
      d[0] = v.x; d[1] = v.y; d[2] = v.z; d[3] = v.w;
    }
    __syncthreads();

    const float* arow = &As[(16 * wave + l16) * PITCH + 2 * hi];
#pragma unroll
    for (int kk = 0; kk < TK; kk += 4) {
      const v2f a = *(const v2f*)(arow + kk);  // A frag: M=l16, K=kk+2hi+{0,1}
#pragma unroll
      for (int j = 0; j < 4; ++j) {
        const float* brow = &Ws[(16 * j + l16) * PITCH + 2 * hi];
        const v2f b = *(const v2f*)(brow + kk);  // B frag: N=l16, same K pair
        acc[j] = wmma_f32(a, b, acc[j]);
      }
    }
    __syncthreads();
  }

  // ---- epilogue: C layout row = r + 8*hi, col = l16 ----
#pragma unroll
  for (int j = 0; j < 4; ++j) {
    const int n  = n0 + 16 * j + l16;
    const float bv = bias[n];
#pragma unroll
    for (int r = 0; r < 8; ++r) {
      Y[(size_t)(m0 + 16 * wave + r + 8 * hi) * HS + n] = acc[j][r] + bv;
    }
  }
}

// ---------------------------------------------------------------------------
// attn_kernel: per (b, h, 64 q-rows). Block = 128 thr (4 waves); each wave
// owns 16 q rows. Three passes over K=2048:
//  1) scores = (q.k^T/8 + rel + 0.05*window)*scale + attn_bias -> d_out raw;
//     K tiles double-buffered in LDS (coalesced float4 staging, bank-clean
//     pitch), row-max via __shfl_xor within the 16-lane half-waves (wave32)
//  2) row sum of exp(s - max) (read-only; L2-resident)
//  3) normalize tiles in the C-layout mapping (coalesced, rmax/rsum already
//     in registers), write final weights, bounce the 16x16 P tile through a
//     per-wave LDS buffer to form A fragments, V tiles double-buffered in
//     LDS -> P.V WMMAs -> ctx
// ---------------------------------------------------------------------------
#define KP 68   // K-tile pitch:  68*l mod 64 = 4*l  -> 16 distinct banks
#define VP 72   // V-tile pitch:  +2 rows shifts banks by 16 (no half clash)
#define PP 20   // P-tile pitch:  20*l mod 64 distinct for 16 lanes
#define NKT (KLEN / 16)

__global__ __launch_bounds__(128) void attn_kernel(
    const float* __restrict__ qp, const float* __restrict__ kp,
    const float* __restrict__ vp, const float* __restrict__ rel_emb,
    const float* __restrict__ attn_bias, const float* __restrict__ log_scale,
    float* __restrict__ wgt_all, float* __restrict__ ctx) {
  __shared__ float Ks[2][16 * KP];
  __shared__ float Vs[2][16 * VP];
  __shared__ float Ps[4][16 * PP];

  const int t    = threadIdx.x;
  const int wave = t >> 5;
  const int lane = t & 31;
  const int l16  = lane & 15;
  const int hi   = lane >> 4;
  const int b    = blockIdx.z;
  const int h    = blockIdx.y;
  const int q0   = blockIdx.x * 64 + wave * 16;
  const int hOff = h * HD;

  float scl = __expf(log_scale[0]);
  scl = fminf(fmaxf(scl, 1.0f), 4.0f);
  const float ab = attn_bias[h];

  float* wgt = wgt_all + (size_t)(b * NH + h) * QLEN * KLEN;

  // cooperative staging geometry: 16 rows x 64 cols, 128 thr x 2 float4
  const int srow = t >> 4;        // 0..7
  const int scol = (t & 15) * 4;  // 0..60  (16 thr cover one 256B row)

  auto stageK = [&](int buf, int kt) {
#pragma unroll
    for (int i = 0; i < 2; ++i) {
      const int row = srow + 8 * i;
      const float4 v = *(const float4*)(
          kp + (size_t)(b * KLEN + kt * 16 + row) * HS + hOff + scol);
      float* d = &Ks[buf][row * KP + scol];
      d[0] = v.x; d[1] = v.y; d[2] = v.z; d[3] = v.w;
    }
  };
  auto stageV = [&](int buf, int kt) {
#pragma unroll
    for (int i = 0; i < 2; ++i) {
      const int row = srow + 8 * i;
      const float4 v = *(const float4*)(
          vp + (size_t)(b * KLEN + kt * 16 + row) * HS + hOff + scol);
      float* d = &Vs[buf][row * VP + scol];
      d[0] = v.x; d[1] = v.y; d[2] = v.z; d[3] = v.w;
    }
  };

  // Preload the wave's 16x64 Q block as 16 A fragments (hd chunks of 4).
  const float* qbase =
      qp + (size_t)(b * QLEN + q0 + l16) * HS + hOff + 2 * hi;
  v2f qa[16];
#pragma unroll
  for (int c4 = 0; c4 < 16; ++c4) qa[c4] = *(const v2f*)(qbase + 4 * c4);

  // ---------------- phase 1: scores + row max ----------------
  float rmax[8];
#pragma unroll
  for (int r = 0; r < 8; ++r) rmax[r] = -3.0e38f;

  stageK(0, 0);
  for (int kt = 0; kt < NKT; ++kt) {
    __syncthreads();
    if (kt + 1 < NKT) stageK((kt + 1) & 1, kt + 1);
    if (kt + 2 < NKT)
      __builtin_prefetch(
          kp + (size_t)(b * KLEN + (kt + 2) * 16 + srow) * HS + hOff + scol,
          0, 1);

    const float* kb = &Ks[kt & 1][l16 * KP + 2 * hi];
    v8f c = (v8f)0.0f;
#pragma unroll
    for (int c4 = 0; c4 < 16; ++c4) {
      const v2f bf = *(const v2f*)(kb + 4 * c4);  // B frag: N=k-idx, K=hd pair
      c = wmma_f32(qa[c4], bf, c);
    }
    const int kcol = kt * 16 + l16;
#pragma unroll
    for (int r = 0; r < 8; ++r) {
      const int qrow = q0 + r + 8 * hi;
      const int dist = kcol - qrow;
      int idx = dist + 511;
      idx = idx < 0 ? 0 : (idx > 1022 ? 1022 : idx);
      const float rel = rel_emb[idx * NH + h];
      const float loc = (dist <= 5 && dist >= -5) ? 0.05f : 0.0f;
      const float s = (c[r] * 0.125f + rel + loc) * scl + ab;
      wgt[(size_t)qrow * KLEN + kcol] = s;  // raw score
      rmax[r] = fmaxf(rmax[r], s);
    }
  }
  // reduce max across the 16 lanes that share each row
#pragma unroll
  for (int r = 0; r < 8; ++r) {
    rmax[r] = fmaxf(rmax[r], __shfl_xor(rmax[r], 1));
    rmax[r] = fmaxf(rmax[r], __shfl_xor(rmax[r], 2));
    rmax[r] = fmaxf(rmax[r], __shfl_xor(rmax[r], 4));
    rmax[r] = fmaxf(rmax[r], __shfl_xor(rmax[r], 8));
  }

  // ---------------- phase 2: row sum of exp ----------------
  float rsum[8];
#pragma unroll
  for (int r = 0; r < 8; ++r) rsum[r] = 0.0f;

  for (int kt = 0; kt < NKT; ++kt) {
#pragma unroll
    for (int r = 0; r < 8; ++r) {
      const size_t off = (size_t)(q0 + r + 8 * hi) * KLEN + kt * 16 + l16;
      rsum[r] += __expf(wgt[off] - rmax[r]);
    }
  }
  float rinv[8];
#pragma unroll
  for (int r = 0; r < 8; ++r) {
    rsum[r] += __shfl_xor(rsum[r], 1);
    rsum[r] += __shfl_xor(rsum[r], 2);
    rsum[r] += __shfl_xor(rsum[r], 4);
    rsum[r] += __shfl_xor(rsum[r], 8);
    rinv[r] = 1.0f / rsum[r];
  }

  // ---------------- phase 3: normalize + write weights + P.V ----------------
  v8f acc[4];
#pragma unroll
  for (int j = 0; j < 4; ++j) acc[j] = (v8f)0.0f;

  float* pw = &Ps[wave][0];
  stageV(0, 0);
  for (int kt = 0; kt < NKT; ++kt) {
    __syncthreads();
    if (kt + 1 < NKT) stageV((kt + 1) & 1, kt + 1);
    if (kt + 2 < NKT)
      __builtin_prefetch(
          vp + (size_t)(b * KLEN + (kt + 2) * 16 + srow) * HS + hOff + scol,
          0, 1);

    // normalize the wave's 16x16 tile in the C-layout mapping (coalesced,
    // rmax/rinv already in registers); final weights out + P tile to LDS
#pragma unroll
    for (int r = 0; r < 8; ++r) {
      const size_t off = (size_t)(q0 + r + 8 * hi) * KLEN + kt * 16 + l16;
      const float p = __expf(wgt[off] - rmax[r]) * rinv[r];
      wgt[off] = p;                       // final normalized weight
      pw[(r + 8 * hi) * PP + l16] = p;    // per-wave LDS bounce
    }
    // A frags from LDS (row = l16), B frags from staged V tile
#pragma unroll
    for (int kc = 0; kc < 4; ++kc) {
      const v2f a = *(const v2f*)&pw[l16 * PP + 4 * kc + 2 * hi];
      const float* vrow = &Vs[kt & 1][(4 * kc + 2 * hi) * VP];
#pragma unroll
      for (int j = 0; j < 4; ++j) {
        v2f bv;
        bv.x = vrow[16 * j + l16];        // V[k,   hdcol]
        bv.y = vrow[VP + 16 * j + l16];   // V[k+1, hdcol]
        acc[j] = wmma_f32(a, bv, acc[j]);
      }
    }
  }

  // context: [B, Q, HS] (head-interleaved along HS)
#pragma unroll
  for (int j = 0; j < 4; ++j) {
#pragma unroll
    for (int r = 0; r < 8; ++r) {
      ctx[(size_t)(b * QLEN + q0 + r + 8 * hi) * HS + hOff + 16 * j + l16] =
          acc[j][r];
    }
  }
}

// ---------------------------------------------------------------------------
// Launch
// ---------------------------------------------------------------------------
extern "C" void kernel_launch(void* const* d_in, const int* in_sizes, int n_in,
                              void* d_out, int out_size, void* d_ws,
                              size_t ws_size, hipStream_t stream) {
  (void)in_sizes; (void)n_in; (void)out_size; (void)ws_size;
  const float* query  = (const float*)d_in[0];
  const float* key    = (const float*)d_in[1];
  const float* value  = (const float*)d_in[2];
  const float* Wq     = (const float*)d_in[3];
  const float* bq     = (const float*)d_in[4];
  const float* Wk     = (const float*)d_in[5];
  const float* bk     = (const float*)d_in[6];
  const float* Wv     = (const float*)d_in[7];
  const float* bv     = (const float*)d_in[8];
  const float* Wo     = (const float*)d_in[9];
  const float* bo     = (const float*)d_in[10];
  const float* rel    = (const float*)d_in[11];
  const float* abias  = (const float*)d_in[12];
  const float* lscale = (const float*)d_in[13];

  // workspace: qp 16MB | kp 32MB | vp 32MB | ctx 16MB  (96 MB total)
  float* qp  = (float*)d_ws;
  float* kp  = qp + (size_t)BSZ * QLEN * HS;
  float* vp  = kp + (size_t)BSZ * KLEN * HS;
  float* ctx = vp + (size_t)BSZ * KLEN * HS;

  float* out = (float*)d_out;                        // [B, Q, HS]
  float* wgt = out + (size_t)BSZ * QLEN * HS;        // [B, NH, Q, K]

  const dim3 blkG(256), blkA(128);
  proj_gemm<<<dim3((BSZ * QLEN) / TM, HS / TN), blkG, 0, stream>>>(query, Wq,
                                                                   bq, qp);
  proj_gemm<<<dim3((BSZ * KLEN) / TM, HS / TN), blkG, 0, stream>>>(key, Wk,
                                                                   bk, kp);
  proj_gemm<<<dim3((BSZ * KLEN) / TM, HS / TN), blkG, 0, stream>>>(value, Wv,
                                                                   bv, vp);
  attn_kernel<<<dim3(QLEN / 64, NH, BSZ), blkA, 0, stream>>>(
      qp, kp, vp, rel, abias, lscale, wgt, ctx);
  proj_gemm<<<dim3((BSZ * QLEN) / TM, HS / TN), blkG, 0, stream>>>(ctx, Wo,
                                                                   bo, out);
}